// PosAttn_TokenOut_4252017623685
// MI455X (gfx1250) — compile-verified
//
#include <hip/hip_runtime.h>
#include <hip/hip_bf16.h>
#include <math.h>

typedef __attribute__((ext_vector_type(2))) float v2f;
typedef __attribute__((ext_vector_type(8))) float v8f;

#define TPB 256

// Reference shapes are fixed module constants (B=4, LSEQ=2048, V=32000, DE=3).
// V is compile-time so all store offsets fold into 24-bit instruction
// immediates (max offset (3*16+7)*32000*4 = 7.04 MB < 2^23).
constexpr int VV = 32000;

__device__ __forceinline__ float softplusf(float x) {
  return fmaxf(x, 0.0f) + log1pf(__expf(-fabsf(x)));
}

__device__ __forceinline__ float waveReduceSum(float v) {
#pragma unroll
  for (int off = 16; off > 0; off >>= 1) v += __shfl_xor(v, off, 32);
  return v;
}

__device__ __forceinline__ float waveReduceMax(float v) {
#pragma unroll
  for (int off = 16; off > 0; off >>= 1) v = fmaxf(v, __shfl_xor(v, off, 32));
  return v;
}

// ---------------------------------------------------------------------------
// Kernel 1: one workgroup per sequence row i.
// scores[b,i,j] depends only on Ppos (M_block touches only positional dims),
// so the softmax weights are batch-independent. Computes weighted sums of
// token embeddings for all 4 batches, applies W[:3,:3] and beta_class, and
// writes hb4[(b*L+i)*4 + d] (d=3 zero-padded for WMMA K=4).
// ---------------------------------------------------------------------------
__global__ void __launch_bounds__(TPB)
row_ctx_kernel(const int* __restrict__ x, const float* __restrict__ E,
               const float* __restrict__ P, const float* __restrict__ M,
               const float* __restrict__ W, const float* __restrict__ rba,
               const float* __restrict__ rbc, float* __restrict__ hb4,
               int L) {
  const int i    = blockIdx.x;
  const int tid  = threadIdx.x;
  const int lane = tid & 31;
  const int wid  = tid >> 5;

  __shared__ float smax[TPB / 32];
  __shared__ float spart[13][TPB / 32];
  __shared__ float sres[13];

  if (i == 0) {
    // fully-masked row -> alpha row is zero -> h = 0
    if (tid < 16) {
      int b = tid >> 2, k = tid & 3;
      hb4[((size_t)b * L) * 4 + k] = 0.0f;
    }
    return;
  }

  const float beta_attn  = softplusf(rba[0]) + 1e-6f;
  const float beta_class = softplusf(rbc[0]) + 1e-6f;

  // score(i,j) = ai*P[j,0] + bi*P[j,1]  (Mp = M[3:5,3:5], row-major 5x5)
  const float Pi0 = P[2 * i], Pi1 = P[2 * i + 1];
  const float ai = beta_attn * (Pi0 * M[18] + Pi1 * M[23]);
  const float bi = beta_attn * (Pi0 * M[19] + Pi1 * M[24]);

  // ---- pass 1: row max over j < i ----
  float lmax = -3.402823466e38f;
  for (int j = tid; j < i; j += TPB)
    lmax = fmaxf(lmax, ai * P[2 * j] + bi * P[2 * j + 1]);
  lmax = waveReduceMax(lmax);
  if (lane == 0) smax[wid] = lmax;
  __syncthreads();
  float rowmax = smax[0];
#pragma unroll
  for (int k = 1; k < TPB / 32; ++k) rowmax = fmaxf(rowmax, smax[k]);

  // ---- pass 2: sum of exp and 4x3 weighted embedding sums ----
  float vals[13];
  {
    float lsum = 0.0f;
    float acc[4][3] = {{0.f,0.f,0.f},{0.f,0.f,0.f},{0.f,0.f,0.f},{0.f,0.f,0.f}};
    for (int j = tid; j < i; j += TPB) {
      const float w = __expf(ai * P[2 * j] + bi * P[2 * j + 1] - rowmax);
      lsum += w;
#pragma unroll
      for (int b = 0; b < 4; ++b) {
        const float* e = E + 3 * (size_t)x[(size_t)b * L + j];
        acc[b][0] += w * e[0];
        acc[b][1] += w * e[1];
        acc[b][2] += w * e[2];
      }
    }
    vals[0] = lsum;
#pragma unroll
    for (int b = 0; b < 4; ++b) {
      vals[1 + b * 3 + 0] = acc[b][0];
      vals[1 + b * 3 + 1] = acc[b][1];
      vals[1 + b * 3 + 2] = acc[b][2];
    }
  }

#pragma unroll
  for (int v = 0; v < 13; ++v) {
    float s = waveReduceSum(vals[v]);
    if (lane == 0) spart[v][wid] = s;
  }
  __syncthreads();
  if (tid < 13) {
    float s = 0.0f;
#pragma unroll
    for (int k = 0; k < TPB / 32; ++k) s += spart[tid][k];
    sres[tid] = s;
  }
  __syncthreads();

  if (tid == 0) {
    const float inv = 1.0f / sres[0];
#pragma unroll
    for (int b = 0; b < 4; ++b) {
      const float c0 = sres[1 + b * 3 + 0] * inv;
      const float c1 = sres[1 + b * 3 + 1] * inv;
      const float c2 = sres[1 + b * 3 + 2] * inv;
      const size_t base = ((size_t)b * L + i) * 4;
      // h = c[0:3] @ W[:3,:3]  (W row-major 5x5), scaled by beta_class
      hb4[base + 0] = beta_class * (c0 * W[0] + c1 * W[5]  + c2 * W[10]);
      hb4[base + 1] = beta_class * (c0 * W[1] + c1 * W[6]  + c2 * W[11]);
      hb4[base + 2] = beta_class * (c0 * W[2] + c1 * W[7]  + c2 * W[12]);
      hb4[base + 3] = 0.0f;  // K=4 padding
    }
  }
}

// ---------------------------------------------------------------------------
// Kernel 2: logits[b,l,v] = hb[b,l,:] . E[v,:]  via V_WMMA_F32_16X16X4_F32.
// Grid: x = column-tile groups (8 waves/block, 1 col tile each),
//       y = row-tile groups (each wave iterates 4 row tiles, reusing the
//           B fragment = E^T tile across all 4 WMMAs).
// No integer division; all store offsets are compile-time immediates.
// A fragment (16x4 f32): V0 = {K0 | K2}, V1 = {K1 | K3} -> one b64 load.
// D: VGPR r -> row (r + 8*hi), col = lane&15.
// ---------------------------------------------------------------------------
__global__ void __launch_bounds__(TPB)
logits_wmma_kernel(const float* __restrict__ hb4, const float* __restrict__ E,
                   float* __restrict__ out) {
  const int lane = threadIdx.x & 31;
  const int wid  = threadIdx.x >> 5;
  const int v0   = (blockIdx.x * (TPB / 32) + wid) << 4;  // column tile
  const int row0 = blockIdx.y << 6;                       // 4 row tiles of 16
  const int m    = lane & 15;
  const int hi   = lane >> 4;

  // B: E^T tile, E row-major [V,3]; K=3 pad = 0. Loaded once, reused 4x.
  const float* e = E + (size_t)(v0 + m) * 3;
  v2f b;
  b.x = e[hi ? 2 : 0];
  b.y = hi ? 0.0f : e[1];

  const float* ap = hb4 + (((size_t)(row0 + m)) << 2) + (hi << 1);
  float* o = out + (size_t)(row0 + (hi << 3)) * VV + v0 + m;

#pragma unroll
  for (int t = 0; t < 4; ++t) {
    const v2f a = *(const v2f*)(ap + (t << 6));  // next 16 rows of hb4
    v8f c = {};
    c = __builtin_amdgcn_wmma_f32_16x16x4_f32(false, a, false, b, (short)0, c,
                                              false, false);
    float* ot = o + t * 16 * VV;
#pragma unroll
    for (int r = 0; r < 8; ++r)
      ot[r * VV] = c[r];  // immediate offsets, 64B-coalesced per half-wave
  }
}

extern "C" void kernel_launch(void* const* d_in, const int* in_sizes, int n_in,
                              void* d_out, int out_size, void* d_ws,
                              size_t ws_size, hipStream_t stream) {
  (void)n_in; (void)out_size; (void)ws_size;
  const int*   x   = (const int*)d_in[0];
  const float* E   = (const float*)d_in[1];
  const float* P   = (const float*)d_in[2];
  const float* M   = (const float*)d_in[3];
  const float* W   = (const float*)d_in[4];
  const float* rba = (const float*)d_in[5];
  const float* rbc = (const float*)d_in[6];

  const int L  = in_sizes[2] / 2;   // Ppos is [L,2]  (2048)
  const int B  = in_sizes[0] / L;   // x is [B,L]     (4)
  const int BL = B * L;             // 8192

  float* hb4 = (float*)d_ws;        // [BL,4] f32 = 128 KB (L2-resident)

  row_ctx_kernel<<<L, TPB, 0, stream>>>(x, E, P, M, W, rba, rbc, hb4, L);

  // x: 32000/16 col tiles / 8 waves = 250 blocks; y: 8192/64 = 128 blocks
  dim3 grid(VV / 16 / (TPB / 32), BL / 64);
  logits_wmma_kernel<<<grid, TPB, 0, stream>>>(hb4, E, (float*)d_out);
}